// context2query_attention_64656437674465
// MI455X (gfx1250) — compile-verified
//
#include <hip/hip_runtime.h>

#define B  64
#define LC 1024
#define LQ 128
#define DD 128

typedef __attribute__((ext_vector_type(2))) float v2f;
typedef __attribute__((ext_vector_type(8))) float v8f;

// ---------------------------------------------------------------------------
// WMMA helper: D = A(16x4) * B(4x16) + C(16x16), fp32.
// Lane layout (wave32): l16 = lane&15, half = lane>>4, koff = 2*half.
//   A: a.x = A[m=l16][koff],  a.y = A[m=l16][koff+1]
//   B: b.x = B[k=koff][n=l16], b.y = B[koff+1][l16]
//   C/D: c[i] = C[m = i + 8*half][n = l16]
// ---------------------------------------------------------------------------
__device__ __forceinline__ v8f wmma_f32(v2f a, v2f b, v8f c) {
  return __builtin_amdgcn_wmma_f32_16x16x4_f32(false, a, false, b, (short)0, c,
                                               false, false);
}

// ---------------------------------------------------------------------------
// Kernel 0: per-row dot with weight vector (s0 = xc@W0, s1 = xq@W1 + bias)
// ---------------------------------------------------------------------------
__global__ void k_rowdot(const float* __restrict__ X, const float* __restrict__ W,
                         const float* __restrict__ bias, float* __restrict__ out,
                         int nrows) {
  int wave = (blockIdx.x * blockDim.x + threadIdx.x) >> 5;
  int lane = threadIdx.x & 31;
  if (wave >= nrows) return;
  const float* row = X + (size_t)wave * DD;
  float a = 0.f;
  for (int k = lane; k < DD; k += 32) a += row[k] * W[k];
  for (int m = 16; m >= 1; m >>= 1) a += __shfl_xor(a, m, 32);
  if (lane == 0) out[wave] = a + (bias ? bias[wave & (LQ - 1)] : 0.f);
}

// ---------------------------------------------------------------------------
// Kernel 1: S[b,c,q] = xc@W0 + xq@W1 + (xc*W2)@xq^T + bias, fused row softmax
// over q (ques_len mask).  Writes raw S and row-softmax P.
// Block = 256 threads = 8 waves; wave w owns c-tile rows [c0, c0+16).
// x_ques[b] staged into LDS pre-scaled by W2 in a K-pair-interleaved layout:
//   element (k,q) at ((k>>1)*LQ + q)*2 + (k&1)
// so the WMMA B operand (k, k+1 at fixed q) is one aligned ds_load_b64.
// ---------------------------------------------------------------------------
__global__ __launch_bounds__(256) void k_scores(
    const float* __restrict__ xc, const float* __restrict__ xq,
    const float* __restrict__ W2, const float* __restrict__ s0c,
    const float* __restrict__ s1qb, const int* __restrict__ qlen,
    float* __restrict__ S, float* __restrict__ P) {
  __shared__ __align__(16) float xqT[DD * LQ];

  const int b = blockIdx.y;
  const int w = threadIdx.x >> 5;
  const int lane = threadIdx.x & 31;
  const int half = lane >> 4;
  const int l16 = lane & 15;
  const int koff = 2 * half;

  const float* xqb = xq + (size_t)b * LQ * DD;
  for (int idx = threadIdx.x; idx < LQ * DD; idx += 256) {
    int q = idx >> 7, k = idx & 127;
    xqT[(((k >> 1) * LQ + q) << 1) + (k & 1)] = xqb[idx] * W2[k];
  }
  __syncthreads();

  const int c0 = (blockIdx.x * 8 + w) * 16;
  const float* xcrow = xc + ((size_t)b * LC + c0 + l16) * DD;

  v8f zero = {0.f, 0.f, 0.f, 0.f, 0.f, 0.f, 0.f, 0.f};
  v8f acc[8];
#pragma unroll
  for (int nt = 0; nt < 8; nt++) acc[nt] = zero;

  for (int k = 0; k < DD; k += 4) {
    v2f a = *(const v2f*)(xcrow + k + koff);        // global_load_b64
    const float* brow = xqT + (((k + koff) >> 1) * LQ) * 2;
#pragma unroll
    for (int nt = 0; nt < 8; nt++) {
      v2f bb = *(const v2f*)(brow + ((nt * 16 + l16) << 1));  // ds_load_b64
      acc[nt] = wmma_f32(a, bb, acc[nt]);
    }
  }

  // epilogue: add scalars, masked row softmax over q (128 cols)
  const int qlenb = qlen[b];
  const float* s1b = s1qb + b * LQ;
  float s1v[8];
#pragma unroll
  for (int nt = 0; nt < 8; nt++) s1v[nt] = s1b[nt * 16 + l16];

  float* Sb = S + ((size_t)b * LC + c0) * LQ;
  float* Pb = P + ((size_t)b * LC + c0) * LQ;

#pragma unroll
  for (int i = 0; i < 8; i++) {
    int m = i + 8 * half;
    float s0v = s0c[b * LC + c0 + m];
    float vals[8], veff[8];
    float mx = -3.0e38f;
#pragma unroll
    for (int nt = 0; nt < 8; nt++) {
      int q = nt * 16 + l16;
      float v = acc[nt][i] + s0v + s1v[nt];
      vals[nt] = v;
      float ve = (q < qlenb) ? v : v - 1.0e12f;
      veff[nt] = ve;
      mx = fmaxf(mx, ve);
    }
    for (int msk = 1; msk < 16; msk <<= 1) mx = fmaxf(mx, __shfl_xor(mx, msk, 32));
    float s = 0.f, e[8];
#pragma unroll
    for (int nt = 0; nt < 8; nt++) {
      e[nt] = expf(veff[nt] - mx);
      s += e[nt];
    }
    for (int msk = 1; msk < 16; msk <<= 1) s += __shfl_xor(s, msk, 32);
    float inv = 1.0f / s;
#pragma unroll
    for (int nt = 0; nt < 8; nt++) {
      int q = nt * 16 + l16;
      Sb[(size_t)m * LQ + q] = vals[nt];
      Pb[(size_t)m * LQ + q] = e[nt] * inv;
    }
  }
}

// ---------------------------------------------------------------------------
// Kernel 2: masked column stats over c (online softmax): colmax, 1/colsum
// ---------------------------------------------------------------------------
__global__ void k_colstats(const float* __restrict__ S, const int* __restrict__ clen,
                           float* __restrict__ colmax, float* __restrict__ colinv) {
  int idx = blockIdx.x * blockDim.x + threadIdx.x;
  if (idx >= B * LQ) return;
  int b = idx >> 7, q = idx & 127;
  int cl = clen[b];
  const float* col = S + (size_t)b * LC * LQ + q;
  float mx = -3.0e38f, s = 0.f;
  for (int c = 0; c < cl; c++) {
    float v = col[(size_t)c * LQ];
    float nm = fmaxf(mx, v);
    s = s * expf(mx - nm) + expf(v - nm);
    mx = nm;
  }
  colmax[idx] = mx;
  colinv[idx] = 1.0f / s;
}

// ---------------------------------------------------------------------------
// Kernel 3: A[b,q,d] = sum_c softmax_c(S)[c,q] * xc[c,d]   (= S_T @ x_cont)
// grid (LQ/16, B), block 64 = 2 waves; wave w owns d-tiles [w*64, w*64+64)
// ---------------------------------------------------------------------------
__global__ __launch_bounds__(64) void k_qc(
    const float* __restrict__ S, const float* __restrict__ xc,
    const float* __restrict__ colmax, const float* __restrict__ colinv,
    const int* __restrict__ clen, float* __restrict__ A) {
  const int b = blockIdx.y;
  const int q0 = blockIdx.x * 16;
  const int w = threadIdx.x >> 5;
  const int lane = threadIdx.x & 31;
  const int half = lane >> 4;
  const int l16 = lane & 15;
  const int koff = 2 * half;
  const int cl = clen[b];

  const float cm = colmax[b * LQ + q0 + l16];
  const float ci = colinv[b * LQ + q0 + l16];
  const float* Sb = S + (size_t)b * LC * LQ;
  const float* xcb = xc + (size_t)b * LC * DD;

  v8f zero = {0.f, 0.f, 0.f, 0.f, 0.f, 0.f, 0.f, 0.f};
  v8f acc[4];
#pragma unroll
  for (int nt = 0; nt < 4; nt++) acc[nt] = zero;

  for (int k = 0; k < LC; k += 4) {
    int cA = k + koff, cB = cA + 1;
    v2f a;
    float v0 = Sb[(size_t)cA * LQ + q0 + l16];
    float v1 = Sb[(size_t)cB * LQ + q0 + l16];
    a.x = (cA < cl) ? expf(v0 - cm) * ci : 0.f;
    a.y = (cB < cl) ? expf(v1 - cm) * ci : 0.f;
#pragma unroll
    for (int nt = 0; nt < 4; nt++) {
      int d0 = w * 64 + nt * 16;
      v2f bb;
      bb.x = xcb[(size_t)cA * DD + d0 + l16];
      bb.y = xcb[(size_t)cB * DD + d0 + l16];
      acc[nt] = wmma_f32(a, bb, acc[nt]);
    }
  }

  float* Ab = A + ((size_t)b * LQ + q0) * DD;
#pragma unroll
  for (int i = 0; i < 8; i++) {
    int m = i + 8 * half;
#pragma unroll
    for (int nt = 0; nt < 4; nt++)
      Ab[(size_t)m * DD + w * 64 + nt * 16 + l16] = acc[nt][i];
  }
}

// ---------------------------------------------------------------------------
// Kernel 4: c2q = P @ xq, q2c = P @ A  (shared A-operand), write 4 segments:
// out[b,c,:] = [xc, c2q, xc*c2q, xc*q2c]
// grid (LC/128, B), block 256 = 8 waves; wave owns one 16-row c-tile.
// ---------------------------------------------------------------------------
__global__ __launch_bounds__(256) void k_out(
    const float* __restrict__ P, const float* __restrict__ xq,
    const float* __restrict__ A, const float* __restrict__ xc,
    float* __restrict__ out) {
  const int b = blockIdx.y;
  const int w = threadIdx.x >> 5;
  const int lane = threadIdx.x & 31;
  const int half = lane >> 4;
  const int l16 = lane & 15;
  const int koff = 2 * half;
  const int c0 = (blockIdx.x * 8 + w) * 16;

  const float* Prow = P + ((size_t)b * LC + c0 + l16) * LQ;
  const float* xqb = xq + (size_t)b * LQ * DD;
  const float* Ab = A + (size_t)b * LQ * DD;

  v8f zero = {0.f, 0.f, 0.f, 0.f, 0.f, 0.f, 0.f, 0.f};
  v8f a1[8], a2[8];
#pragma unroll
  for (int nt = 0; nt < 8; nt++) { a1[nt] = zero; a2[nt] = zero; }

  for (int k = 0; k < LQ; k += 4) {
    v2f a = *(const v2f*)(Prow + k + koff);        // global_load_b64
#pragma unroll
    for (int nt = 0; nt < 8; nt++) {
      int d0 = nt * 16 + l16;
      v2f bq, bA;
      bq.x = xqb[(size_t)(k + koff) * DD + d0];
      bq.y = xqb[(size_t)(k + koff + 1) * DD + d0];
      a1[nt] = wmma_f32(a, bq, a1[nt]);
      bA.x = Ab[(size_t)(k + koff) * DD + d0];
      bA.y = Ab[(size_t)(k + koff + 1) * DD + d0];
      a2[nt] = wmma_f32(a, bA, a2[nt]);
    }
  }

  const float* xcb = xc + (size_t)b * LC * DD;
  float* ob = out + (size_t)b * LC * (4 * DD);
#pragma unroll
  for (int i = 0; i < 8; i++) {
    int c = c0 + i + 8 * half;
    const float* xr = xcb + (size_t)c * DD;
    float* o = ob + (size_t)c * (4 * DD);
#pragma unroll
    for (int nt = 0; nt < 8; nt++) {
      int d = nt * 16 + l16;
      float xv = xr[d];
      float c2 = a1[nt][i];
      float q2 = a2[nt][i];
      o[d] = xv;
      o[DD + d] = c2;
      o[2 * DD + d] = xv * c2;
      o[3 * DD + d] = xv * q2;
    }
  }
}

// ---------------------------------------------------------------------------
extern "C" void kernel_launch(void* const* d_in, const int* in_sizes, int n_in,
                              void* d_out, int out_size, void* d_ws, size_t ws_size,
                              hipStream_t stream) {
  const float* xc   = (const float*)d_in[0];
  const float* xq   = (const float*)d_in[1];
  const float* W0   = (const float*)d_in[2];
  const float* W1   = (const float*)d_in[3];
  const float* W2   = (const float*)d_in[4];
  const float* bias = (const float*)d_in[5];
  const int* clen   = (const int*)d_in[6];
  const int* qlen   = (const int*)d_in[7];
  float* out        = (float*)d_out;

  float* ws = (float*)d_ws;
  float* s0c    = ws;                                // B*LC
  float* s1qb   = s0c + B * LC;                      // B*LQ
  float* colmax = s1qb + B * LQ;                     // B*LQ
  float* colinv = colmax + B * LQ;                   // B*LQ
  float* Abuf   = colinv + B * LQ;                   // B*LQ*DD
  float* Sbuf   = Abuf + (size_t)B * LQ * DD;        // B*LC*LQ
  float* Pbuf   = Sbuf + (size_t)B * LC * LQ;        // B*LC*LQ

  k_rowdot<<<(B * LC * 32 + 255) / 256, 256, 0, stream>>>(xc, W0, nullptr, s0c, B * LC);
  k_rowdot<<<(B * LQ * 32 + 255) / 256, 256, 0, stream>>>(xq, W1, bias, s1qb, B * LQ);

  dim3 g1(LC / 128, B);
  k_scores<<<g1, 256, 0, stream>>>(xc, xq, W2, s0c, s1qb, qlen, Sbuf, Pbuf);

  k_colstats<<<(B * LQ + 255) / 256, 256, 0, stream>>>(Sbuf, clen, colmax, colinv);

  dim3 g3(LQ / 16, B);
  k_qc<<<g3, 64, 0, stream>>>(Sbuf, xc, colmax, colinv, clen, Abuf);

  dim3 g4(LC / 128, B);
  k_out<<<g4, 256, 0, stream>>>(Pbuf, xq, Abuf, xc, out);
}